// Capsule_44779329028201
// MI455X (gfx1250) — compile-verified
//
#include <hip/hip_runtime.h>
#include <math.h>

// Capsule dynamic routing, algebraically refactored so u_hat is never
// materialized. Three passes over u_vecs (256 MB each) instead of ~3 GB.
//
//   U[b,i]    = sum_n u[b,n,i]                       (kernel caps_colsum)
//   S0        = U/32 (uniform softmax of b=0)
//   out,t     = cal/normalize/t-update from S,W      (kernel caps_update)
//   S[b,k,i]  = sum_n softmax_k(u[b,n,:]·t[b,k,:]) * u[b,n,i]
//                                                    (kernel caps_routing_pass, WMMA f32)
// repeated for routing iterations 1 and 2.

typedef float v2f __attribute__((ext_vector_type(2)));
typedef float v8f __attribute__((ext_vector_type(8)));

#define B_BATCH   64
#define N_ROWS    4096
#define D_IN      256
#define NCAP      32
#define DCAP      16
#define NCHUNK    8            // n-chunks per batch for routing pass
#define SUBTILES  8            // 512 rows / 64 rows per sub-tile
#define TROWS     64
#define UPITCH    260          // padded LDS pitch (floats) for u tile rows
#define TPITCH    260          // padded LDS pitch for t rows
#define CPITCH    66           // padded pitch for transposed c
#define LPITCH    33           // padded pitch for raw logits
#define ACH       16           // n-chunks for column-sum kernel

// ---------------------------------------------------------------- kernel A
__global__ void caps_colsum(const float* __restrict__ u, float* __restrict__ U_part) {
    const int b  = blockIdx.x;
    const int ch = blockIdx.y;           // 0..15, 256 rows each
    const int i  = threadIdx.x;          // 0..255
    const float* p = u + ((size_t)b * N_ROWS + (size_t)ch * 256) * D_IN + i;
    float acc = 0.0f;
    for (int n = 0; n < 256; ++n) acc += p[(size_t)n * D_IN];
    U_part[((size_t)ch * B_BATCH + b) * D_IN + i] = acc;
}

// ---------------------------------------------------------------- kernel B
// One block = (batch b, chunk of 512 n-rows). 256 threads = 8 waves.
// Accumulates partial S[b, 32, 256] for its chunk, stored to S_part[chunk].
__global__ void caps_routing_pass(const float* __restrict__ u,   // [64][4096][256]
                                  const float* __restrict__ t,   // [64][32][256]
                                  float* __restrict__ S_part) {  // [NCHUNK][64][32][256]
    extern __shared__ float smem[];
    float* u_tile = smem;                       // [64][UPITCH]
    float* t_lds  = u_tile + TROWS * UPITCH;    // [32][TPITCH]
    float* cT     = t_lds + NCAP * TPITCH;      // [32][CPITCH]  (c transposed)
    float* lin    = cT + NCAP * CPITCH;         // [64][LPITCH]  (raw logits)

    const int b     = blockIdx.x;
    const int chunk = blockIdx.y;
    const int tid   = threadIdx.x;
    const int w     = tid >> 5;
    const int lane  = tid & 31;
    const int lh    = lane >> 4;     // half-wave select (K-pair)
    const int ll    = lane & 15;

    // stage t[b] (32x256) into LDS with padded pitch
    {
        const float4* tg = (const float4*)(t + (size_t)b * NCAP * D_IN);
        for (int r = 0; r < 8; ++r) {
            int idx = tid + r * 256;             // float4 index 0..2047
            int row = idx >> 6;                  // /64 float4 per row
            int c4  = idx & 63;
            float4 v = tg[idx];
            float* dst = t_lds + row * TPITCH + c4 * 4;
            dst[0] = v.x; dst[1] = v.y; dst[2] = v.z; dst[3] = v.w;
        }
    }

    // phase-2 persistent accumulators: wave owns captile = w&1, 4 i-tiles
    v8f acc[4];
    acc[0] = v8f{}; acc[1] = v8f{}; acc[2] = v8f{}; acc[3] = v8f{};
    const int captile = w & 1;
    const int ibase0  = (w >> 1) * 4;            // first i-tile index (of 16)

    const int n_chunk0 = chunk * (N_ROWS / NCHUNK);

    for (int st = 0; st < SUBTILES; ++st) {
        const int n0 = n_chunk0 + st * TROWS;
        __syncthreads();  // prior phase-2 reads of u_tile complete
        // stage u[b, n0:n0+64, :] into LDS (coalesced float4)
        const float4* ug = (const float4*)(u + ((size_t)b * N_ROWS + n0) * D_IN);
        for (int r = 0; r < 16; ++r) {
            int idx = tid + r * 256;             // float4 index 0..4095
            int row = idx >> 6;
            int c4  = idx & 63;
            float4 v = ug[idx];
            float* dst = u_tile + row * UPITCH + c4 * 4;
            dst[0] = v.x; dst[1] = v.y; dst[2] = v.z; dst[3] = v.w;
        }
        __syncthreads();

        // ---- phase 1: logits[64x32] = u_tile @ t^T  (WMMA f32 16x16x4) ----
        {
            const int mt = w >> 1;               // n-tile 0..3
            const int ct = w & 1;                // cap-tile 0..1
            v8f d = v8f{};
            const float* aBase = u_tile + (mt * 16 + ll) * UPITCH + 2 * lh;
            const float* bBase = t_lds + (ct * 16 + ll) * TPITCH + 2 * lh;
            for (int k = 0; k < D_IN / 4; ++k) {
                v2f a  = *(const v2f*)(aBase + 4 * k);
                v2f bb = *(const v2f*)(bBase + 4 * k);
                d = __builtin_amdgcn_wmma_f32_16x16x4_f32(
                        false, a, false, bb, (short)0, d, false, false);
            }
            for (int r = 0; r < 8; ++r) {
                int n = mt * 16 + (lh ? r + 8 : r);
                int c = ct * 16 + ll;
                lin[n * LPITCH + c] = d[r];
            }
        }
        __syncthreads();

        // ---- softmax over 32 capsules per row; write transposed ----
        if (tid < TROWS) {
            float row[NCAP];
            float m = -3.0e38f;
            for (int c = 0; c < NCAP; ++c) {
                row[c] = lin[tid * LPITCH + c];
                m = fmaxf(m, row[c]);
            }
            float s = 0.0f;
            for (int c = 0; c < NCAP; ++c) { row[c] = __expf(row[c] - m); s += row[c]; }
            float inv = 1.0f / s;
            for (int c = 0; c < NCAP; ++c) cT[c * CPITCH + tid] = row[c] * inv;
        }
        __syncthreads();

        // ---- phase 2: S[32x256] += cT(32x64) @ u_tile(64x256) ----
        {
            const float* aBase = cT + (captile * 16 + ll) * CPITCH + 2 * lh;
            for (int kt = 0; kt < TROWS / 4; ++kt) {
                const int nb = kt * 4;
                v2f a = *(const v2f*)(aBase + nb);
                const int rrow = nb + 2 * lh;
                for (int j = 0; j < 4; ++j) {
                    const int i0 = (ibase0 + j) * 16;
                    v2f bb;
                    bb.x = u_tile[rrow * UPITCH + i0 + ll];
                    bb.y = u_tile[(rrow + 1) * UPITCH + i0 + ll];
                    acc[j] = __builtin_amdgcn_wmma_f32_16x16x4_f32(
                                 false, a, false, bb, (short)0, acc[j], false, false);
                }
            }
        }
    }

    // store this chunk's partial S (plain stores -> deterministic)
    float* sp = S_part + (((size_t)chunk * B_BATCH + b) * NCAP) * D_IN;
    for (int j = 0; j < 4; ++j) {
        const int i0 = (ibase0 + j) * 16;
        for (int r = 0; r < 8; ++r) {
            int cap = captile * 16 + (lh ? r + 8 : r);
            sp[(size_t)cap * D_IN + i0 + ll] = acc[j][r];
        }
    }
}

// ---------------------------------------------------------------- kernel C
// Per (b,k): cal[d] = sum_i s_i * W[i, k*16+d]; out = cal/||cal||;
// t[b,k,i] = sum_d W[i,k*16+d] * out[d].
__global__ void caps_update(const float* __restrict__ SorU,  // mode0: U_part, mode1: S_part
                            const float* __restrict__ W,     // [256][512]
                            float* __restrict__ out,         // [64][32][16]
                            float* __restrict__ t,           // [64][32][256]
                            int mode) {
    const int b = blockIdx.x;
    const int k = blockIdx.y;
    const int i = threadIdx.x;   // 0..255
    __shared__ float red[256][DCAP + 1];
    __shared__ float outs[DCAP];

    float s;
    if (mode == 0) {
        float acc = 0.0f;
        for (int ch = 0; ch < ACH; ++ch)
            acc += SorU[((size_t)ch * B_BATCH + b) * D_IN + i];
        s = acc * (1.0f / (float)NCAP);          // uniform softmax of b=0
    } else {
        float acc = 0.0f;
        for (int ch = 0; ch < NCHUNK; ++ch)
            acc += SorU[(((size_t)ch * B_BATCH + b) * NCAP + k) * D_IN + i];
        s = acc;
    }

    float wrow[DCAP];
    const float* wp = W + (size_t)i * (NCAP * DCAP) + k * DCAP;
    for (int d = 0; d < DCAP; ++d) wrow[d] = wp[d];

    for (int d = 0; d < DCAP; ++d) red[i][d] = s * wrow[d];
    __syncthreads();
    for (int off = 128; off >= 1; off >>= 1) {
        if (i < off)
            for (int d = 0; d < DCAP; ++d) red[i][d] += red[i + off][d];
        __syncthreads();
    }
    if (i == 0) {
        float ss = 0.0f;
        for (int d = 0; d < DCAP; ++d) ss += red[0][d] * red[0][d];
        float inv = 1.0f / sqrtf(ss + 1e-7f);
        for (int d = 0; d < DCAP; ++d) {
            float o = red[0][d] * inv;
            outs[d] = o;
            out[(((size_t)b * NCAP) + k) * DCAP + d] = o;
        }
    }
    __syncthreads();
    float tv = 0.0f;
    for (int d = 0; d < DCAP; ++d) tv += wrow[d] * outs[d];
    t[(((size_t)b * NCAP) + k) * D_IN + i] = tv;
}

// ---------------------------------------------------------------- launch
extern "C" void kernel_launch(void* const* d_in, const int* in_sizes, int n_in,
                              void* d_out, int out_size, void* d_ws, size_t ws_size,
                              hipStream_t stream) {
    const float* u = (const float*)d_in[0];   // [64,4096,256]
    const float* W = (const float*)d_in[1];   // [1,256,512]
    float* outp = (float*)d_out;              // [64,32,16]

    float* ws     = (float*)d_ws;
    float* U_part = ws;                                        // 16*64*256
    float* S_part = U_part + (size_t)ACH * B_BATCH * D_IN;     // 8*64*32*256
    float* t_buf  = S_part + (size_t)NCHUNK * B_BATCH * NCAP * D_IN;  // 64*32*256

    const size_t smem = (size_t)(TROWS * UPITCH + NCAP * TPITCH +
                                 NCAP * CPITCH + TROWS * LPITCH) * sizeof(float);

    // iteration 0 (uniform coupling)
    caps_colsum<<<dim3(B_BATCH, ACH), 256, 0, stream>>>(u, U_part);
    caps_update<<<dim3(B_BATCH, NCAP), 256, 0, stream>>>(U_part, W, outp, t_buf, 0);
    // iteration 1
    caps_routing_pass<<<dim3(B_BATCH, NCHUNK), 256, smem, stream>>>(u, t_buf, S_part);
    caps_update<<<dim3(B_BATCH, NCAP), 256, 0, stream>>>(S_part, W, outp, t_buf, 1);
    // iteration 2 (final)
    caps_routing_pass<<<dim3(B_BATCH, NCHUNK), 256, smem, stream>>>(u, t_buf, S_part);
    caps_update<<<dim3(B_BATCH, NCAP), 256, 0, stream>>>(S_part, W, outp, t_buf, 1);
}